// SparseSpatial2Channel_12077448036742
// MI455X (gfx1250) — compile-verified
//
#include <hip/hip_runtime.h>

typedef float    f4 __attribute__((ext_vector_type(4)));
typedef int      i4 __attribute__((ext_vector_type(4)));
typedef unsigned u4 __attribute__((ext_vector_type(4)));

#define NCODES  524288          // B * 64^3 possible coarse-cell codes
#define SCAN_T  256             // threads per scan block
#define EPB     1024            // bitmap elements per scan block (4 per thread)
#define NSCANB  (NCODES / EPB)  // 512 scan blocks
#define SENTINEL 0xFFFFFFFFu

#if defined(__has_builtin)
#  if __has_builtin(__builtin_amdgcn_global_load_async_to_lds_b128)
#    define HAVE_ASYNC_LDS 1
#  endif
#  if __has_builtin(__builtin_amdgcn_s_wait_asynccnt)
#    define HAVE_WAIT_ASYNC 1
#  endif
#endif

#ifdef HAVE_ASYNC_LDS
typedef __attribute__((address_space(1))) i4 as1_i4;   // global (device) v4i32
typedef __attribute__((address_space(3))) i4 as3_i4;   // LDS v4i32
#endif

// ---------------- init kernels ----------------

__global__ void k_fill_u4(u4* __restrict__ p, unsigned val, int n4) {
    int g = blockIdx.x * blockDim.x + threadIdx.x;
    if (g < n4) {
        u4 v = {val, val, val, val};
        p[g] = v;
    }
}

__global__ void k_fill_coords(f4* __restrict__ coords4, int n) {
    int g = blockIdx.x * blockDim.x + threadIdx.x;
    if (g < n) {
        // decode(fill_value=524288) = (2,0,0,0)
        f4 v = {2.f, 0.f, 0.f, 0.f};
        __builtin_nontemporal_store(v, &coords4[g]);
    }
}

// ---------------- mark: presence bitmap + inverse map (unique writer per slot) ----

__global__ void k_mark(const i4* __restrict__ coords,
                       unsigned* __restrict__ present,
                       unsigned* __restrict__ cell2voxel, int n) {
    int g = blockIdx.x * blockDim.x + threadIdx.x;
    if (g >= n) return;
    i4 c = coords[g];   // (b, x, y, z)
    unsigned code = ((unsigned)c.x << 18) | (((unsigned)c.y >> 1) << 12)
                  | (((unsigned)c.z >> 1) << 6) | ((unsigned)c.w >> 1);
    unsigned sub = ((unsigned)c.y & 1u) | (((unsigned)c.z & 1u) << 1) | (((unsigned)c.w & 1u) << 2);
    present[code] = 1u;                      // benign same-value race
    cell2voxel[code * 8 + sub] = (unsigned)g + 1u;  // unique writer: voxels are distinct
}

// ---------------- two-level exclusive scan over 512K flags ----------------

__global__ void k_block_sums(const u4* __restrict__ present4, unsigned* __restrict__ blockSums) {
    __shared__ unsigned sh[SCAN_T];
    int t = threadIdx.x;
    u4 v = present4[blockIdx.x * (EPB / 4) + t];
    unsigned tsum = v.x + v.y + v.z + v.w;
    sh[t] = tsum;
    __syncthreads();
    for (int off = 1; off < SCAN_T; off <<= 1) {
        unsigned a = (t >= off) ? sh[t - off] : 0u;
        __syncthreads();
        sh[t] += a;
        __syncthreads();
    }
    if (t == SCAN_T - 1) blockSums[blockIdx.x] = sh[t];
}

__global__ void k_scan_offsets(unsigned* __restrict__ blockSums) {
    __shared__ unsigned sh[NSCANB];
    int t = threadIdx.x;          // launched with NSCANB=512 threads, 1 block
    unsigned v = blockSums[t];
    sh[t] = v;
    __syncthreads();
    for (int off = 1; off < NSCANB; off <<= 1) {
        unsigned a = (t >= off) ? sh[t - off] : 0u;
        __syncthreads();
        sh[t] += a;
        __syncthreads();
    }
    blockSums[t] = sh[t] - v;     // exclusive block offset
}

__global__ void k_scan_rows(const u4* __restrict__ present4,
                            const unsigned* __restrict__ blockOffsets,
                            unsigned* __restrict__ rowcode,
                            f4* __restrict__ coordsOut4) {
    __shared__ unsigned sh[SCAN_T];
    int t = threadIdx.x;
    u4 v = present4[blockIdx.x * (EPB / 4) + t];
    unsigned tsum = v.x + v.y + v.z + v.w;
    sh[t] = tsum;
    __syncthreads();
    for (int off = 1; off < SCAN_T; off <<= 1) {
        unsigned a = (t >= off) ? sh[t - off] : 0u;
        __syncthreads();
        sh[t] += a;
        __syncthreads();
    }
    unsigned excl = blockOffsets[blockIdx.x] + sh[t] - tsum;

    unsigned base = blockIdx.x * EPB + t * 4;
    unsigned r = excl;
    #pragma unroll
    for (int j = 0; j < 4; ++j) {
        unsigned pj = (j == 0) ? v.x : (j == 1) ? v.y : (j == 2) ? v.z : v.w;
        if (pj) {
            unsigned code = base + j;
            rowcode[r] = code;    // rank -> code (sorted ascending by construction)
            f4 cv = {(float)(code >> 18), (float)((code >> 12) & 63u),
                     (float)((code >> 6) & 63u), (float)(code & 63u)};
            coordsOut4[r] = cv;
            r += 1u;
        }
    }
}

// -------- row writer: each 1KB output row written exactly once, contiguously ------

__global__ void k_rows(const f4* __restrict__ feats4,
                       const unsigned* __restrict__ rowcode,
                       const unsigned* __restrict__ cell2voxel,
                       f4* __restrict__ out4, int n) {
#ifdef HAVE_ASYNC_LDS
    __shared__ f4 stage[256];
#endif
    int g = blockIdx.x * blockDim.x + threadIdx.x;  // one float4 of output per thread
    bool active = g < n * 64;                        // 64 float4 per row
    int r = g >> 6;                                  // output row
    int p = g & 63;                                  // float4 index within row
    unsigned vi = 0u;
    if (active) {
        unsigned code = rowcode[r];
        if (code != SENTINEL) {
            int sub  = p >> 3;                       // which of 8 children
            int part = p & 7;                        // which float4 of that child
            vi = cell2voxel[code * 8 + sub];
            if (vi) {
                f4* src = const_cast<f4*>(&feats4[(size_t)(vi - 1u) * 8 + part]);
#ifdef HAVE_ASYNC_LDS
                // CDNA5 async DMA: global -> LDS, tracked by ASYNCcnt
                __builtin_amdgcn_global_load_async_to_lds_b128(
                    (as1_i4*)src, (as3_i4*)&stage[threadIdx.x], 0, 0);
#else
                f4 tv = *src;
                __builtin_nontemporal_store(tv, &out4[g]);
#endif
            }
        }
    }
#ifdef HAVE_ASYNC_LDS
#  ifdef HAVE_WAIT_ASYNC
    __builtin_amdgcn_s_wait_asynccnt(0);
#  else
    asm volatile("s_wait_asynccnt 0x0" ::: "memory");
#  endif
    if (active) {
        f4 v = {0.f, 0.f, 0.f, 0.f};
        if (vi) v = stage[threadIdx.x];
        __builtin_nontemporal_store(v, &out4[g]);   // fully linear NT b128 stream
    }
#else
    if (active && !vi) {
        f4 z = {0.f, 0.f, 0.f, 0.f};
        __builtin_nontemporal_store(z, &out4[g]);
    }
#endif
}

// ---------------- launch ----------------

extern "C" void kernel_launch(void* const* d_in, const int* in_sizes, int n_in,
                              void* d_out, int out_size, void* d_ws, size_t ws_size,
                              hipStream_t stream) {
    const float* feats  = (const float*)d_in[0];
    const int*   coords = (const int*)d_in[1];
    float*       out    = (float*)d_out;

    const int N = in_sizes[0] / 32;                  // C = 32
    float* featsOut  = out;                          // [N, 256] floats
    float* coordsOut = out + (size_t)N * 256;        // [N, 4] coord values as floats

    unsigned* present    = (unsigned*)d_ws;          // 524288 u32  (2 MB)
    unsigned* cell2voxel = present + NCODES;         // 4194304 u32 (16 MB)
    unsigned* rowcode    = cell2voxel + NCODES * 8;  // N u32      (1 MB)
    unsigned* blockSums  = rowcode + N;              // 512 u32

    k_fill_u4     <<<(NCODES / 4 + 255) / 256, 256, 0, stream>>>((u4*)present, 0u, NCODES / 4);
    k_fill_u4     <<<(NCODES * 2 + 255) / 256, 256, 0, stream>>>((u4*)cell2voxel, 0u, NCODES * 2);
    k_fill_u4     <<<(N / 4 + 255) / 256, 256, 0, stream>>>((u4*)rowcode, SENTINEL, N / 4);
    k_fill_coords <<<(N + 255) / 256, 256, 0, stream>>>((f4*)coordsOut, N);
    k_mark        <<<(N + 255) / 256, 256, 0, stream>>>((const i4*)coords, present, cell2voxel, N);
    k_block_sums  <<<NSCANB, SCAN_T, 0, stream>>>((const u4*)present, blockSums);
    k_scan_offsets<<<1, NSCANB, 0, stream>>>(blockSums);
    k_scan_rows   <<<NSCANB, SCAN_T, 0, stream>>>((const u4*)present, blockSums, rowcode, (f4*)coordsOut);
    k_rows        <<<(N * 64 + 255) / 256, 256, 0, stream>>>((const f4*)feats, rowcode, cell2voxel,
                                                             (f4*)featsOut, N);
}